// CausalLinearAttention_18915035971883
// MI455X (gfx1250) — compile-verified
//
#include <hip/hip_runtime.h>

typedef __attribute__((ext_vector_type(2))) float v2f;
typedef __attribute__((ext_vector_type(8))) float v8f;

#define B_  4
#define L_  2048
#define H_  8
#define D_  64
#define C_  64
#define NC_ (L_ / C_)        // 32 chunks
#define BH_ (B_ * H_)        // 32
#define PAD 68               // LDS row stride (floats), breaks bank conflicts
#define EPS_ 1e-10f

// feature map: elu(x)+1  ==  x+1 (x>0) else exp(x)
__device__ __forceinline__ float fmap(float x) {
    return x > 0.0f ? x + 1.0f : __expf(x);
}

// D(16x16,f32) += A(16x4,f32) * B(4x16,f32)
__device__ __forceinline__ v8f wmma_k4(v2f a, v2f b, v8f c) {
    return __builtin_amdgcn_wmma_f32_16x16x4_f32(
        /*neg_a=*/false, a, /*neg_b=*/false, b,
        /*c_mod=*/(short)0, c, /*reuse_a=*/false, /*reuse_b=*/false);
}

// ---- CDNA5 async global->LDS path (ASYNCcnt), GVS addressing mode ----
// LDS generic pointers keep the LDS byte offset in the low 32 bits,
// so truncation yields the DS address.
__device__ __forceinline__ unsigned lds_off(const void* p) {
    return (unsigned)(uintptr_t)p;
}
// global_load_async_to_lds_b128: VDST = LDS byte addr VGPR, VADDR = 32-bit
// byte offset VGPR, SADDR = 64-bit SGPR base.  16 bytes per lane.
__device__ __forceinline__ void async_load_b128(unsigned lds_addr, unsigned goff,
                                                const void* sbase) {
    asm volatile("global_load_async_to_lds_b128 %0, %1, %2"
                 :: "v"(lds_addr), "v"(goff), "s"(sbase)
                 : "memory");
}
__device__ __forceinline__ void wait_async0() {
    asm volatile("s_wait_asynccnt 0x0" ::: "memory");
}

// ---------------------------------------------------------------------------
// Phase 1: per-chunk KV outer-product sums  KVc[d][m] = sum_l fm(K)[l][d]*V[l][m]
// and per-chunk key sums Ksumc[d]. One block per (b,h,chunk); 16 waves, one
// 16x16 tile of the 64x64 KV matrix per wave via 16 chained f32 WMMAs.
// V is staged with async global->LDS; K goes through VGPRs for the feature map.
// ---------------------------------------------------------------------------
__global__ __launch_bounds__(512)
void kv_chunk_kernel(const float* __restrict__ Kin, const float* __restrict__ Vin,
                     float* __restrict__ kvbuf, float* __restrict__ ksumbuf) {
    __shared__ float Kf[C_][PAD];
    __shared__ float Vv[C_][PAD];
    const int chunk = blockIdx.x, h = blockIdx.y, b = blockIdx.z;
    const int bh = b * H_ + h;
    const int tid = threadIdx.x;
    const int lane = tid & 31, wave = tid >> 5;
    const size_t base = (((size_t)b * L_ + (size_t)chunk * C_) * H_ + h) * D_;

    // async: V chunk -> LDS (1024 x 16B granules over 512 threads)
    const float* vbase = Vin + base;
    for (int i = 0; i < 2; ++i) {
        int e = tid + i * 512;
        int r = e >> 4, q = e & 15;
        unsigned goff = (unsigned)((r * (H_ * D_) + q * 4) * 4);
        async_load_b128(lds_off(&Vv[r][q * 4]), goff, (const void*)vbase);
    }
    // VGPR path: K chunk with feature map
    for (int i = 0; i < 8; ++i) {
        int idx = tid + i * 512;
        int r = idx >> 6, d = idx & 63;
        Kf[r][d] = fmap(Kin[base + (size_t)r * (H_ * D_) + d]);
    }
    wait_async0();
    __syncthreads();

    const int tm = wave >> 2, tn = wave & 3;
    const int lm = lane & 15, lh = lane >> 4;
    v8f acc = {};
    for (int kk = 0; kk < 16; ++kk) {
        int kb = kk * 4 + lh * 2;          // K index base for this lane-half
        v2f a, bf;
        // A(M=d,K=l) = Kf[l][d]  (transposed read)
        a.x = Kf[kb + 0][tm * 16 + lm];
        a.y = Kf[kb + 1][tm * 16 + lm];
        // B(K=l,N=m) = Vv[l][m]
        bf.x = Vv[kb + 0][tn * 16 + lm];
        bf.y = Vv[kb + 1][tn * 16 + lm];
        acc = wmma_k4(a, bf, acc);
    }
    float* kvout = kvbuf + (size_t)(bh * NC_ + chunk) * (D_ * D_);
    for (int r = 0; r < 8; ++r) {
        int d = tm * 16 + r + lh * 8;
        int m = tn * 16 + lm;
        kvout[d * D_ + m] = acc[r];
    }
    if (tid < D_) {
        float s = 0.0f;
        for (int r = 0; r < C_; ++r) s += Kf[r][tid];
        ksumbuf[(size_t)(bh * NC_ + chunk) * D_ + tid] = s;
    }
}

// ---------------------------------------------------------------------------
// Phase 2: in-place exclusive prefix over chunks for KV (4096 f32 per chunk)
// and key sums (64 f32 per chunk). One block per (b,h); each thread owns a
// disjoint slice of elements, so in-place is race-free.
// ---------------------------------------------------------------------------
__global__ __launch_bounds__(256)
void prefix_kernel(float* __restrict__ kvbuf, float* __restrict__ ksumbuf) {
    const int bh = blockIdx.x;
    const int tid = threadIdx.x;
    float acc[16];
    for (int i = 0; i < 16; ++i) acc[i] = 0.0f;
    for (int c = 0; c < NC_; ++c) {
        size_t base = (size_t)(bh * NC_ + c) * (D_ * D_) + (size_t)tid * 16;
        for (int i = 0; i < 16; ++i) {
            float t = kvbuf[base + i];
            kvbuf[base + i] = acc[i];
            acc[i] += t;
        }
    }
    if (tid < D_) {
        float a2 = 0.0f;
        for (int c = 0; c < NC_; ++c) {
            size_t base = (size_t)(bh * NC_ + c) * D_ + tid;
            float t = ksumbuf[base];
            ksumbuf[base] = a2;
            a2 += t;
        }
    }
}

// ---------------------------------------------------------------------------
// Phase 3: per (b,h,chunk):  A = mask(Q'K'^T);  O = A*V + Q'*KV_prev;
// denom = rowsum(A) + Q'.Ksum_prev + eps;  out = O / denom.
// 48 chained f32 WMMAs per wave.  V and KV_prev staged via async->LDS.
// ---------------------------------------------------------------------------
__global__ __launch_bounds__(512)
void out_kernel(const float* __restrict__ Qin, const float* __restrict__ Kin,
                const float* __restrict__ Vin, const float* __restrict__ kvbuf,
                const float* __restrict__ ksumbuf, float* __restrict__ Out) {
    __shared__ float Qf[C_][PAD];
    __shared__ float Kf[C_][PAD];
    __shared__ float Vv[C_][PAD];
    __shared__ float Asc[C_][PAD];
    __shared__ float KVp[D_][PAD];
    __shared__ float ksum[D_];
    __shared__ float invden[C_];
    const int chunk = blockIdx.x, h = blockIdx.y, b = blockIdx.z;
    const int bh = b * H_ + h;
    const int tid = threadIdx.x;
    const int lane = tid & 31, wave = tid >> 5;
    const size_t base = (((size_t)b * L_ + (size_t)chunk * C_) * H_ + h) * D_;
    const float* kvsrc = kvbuf + (size_t)(bh * NC_ + chunk) * (D_ * D_);

    // async: V chunk and KV_prev -> LDS (16B granules)
    const float* vbase = Vin + base;
    for (int i = 0; i < 2; ++i) {
        int e = tid + i * 512;
        int r = e >> 4, q = e & 15;
        unsigned goffV = (unsigned)((r * (H_ * D_) + q * 4) * 4);
        async_load_b128(lds_off(&Vv[r][q * 4]), goffV, (const void*)vbase);
        unsigned goffKV = (unsigned)(e * 16);           // contiguous 64x64 f32
        async_load_b128(lds_off(&KVp[r][q * 4]), goffKV, (const void*)kvsrc);
    }
    // VGPR path: Q and K with feature map
    for (int i = 0; i < 8; ++i) {
        int idx = tid + i * 512;
        int r = idx >> 6, d = idx & 63;
        size_t g = base + (size_t)r * (H_ * D_) + d;
        Qf[r][d] = fmap(Qin[g]);
        Kf[r][d] = fmap(Kin[g]);
    }
    if (tid < D_) ksum[tid] = ksumbuf[(size_t)(bh * NC_ + chunk) * D_ + tid];
    wait_async0();
    __syncthreads();

    const int tm = wave >> 2, tn = wave & 3;
    const int lm = lane & 15, lh = lane >> 4;

    // ---- A = Q' * K'^T (16 WMMAs), causal mask on store ----
    {
        v8f acc = {};
        for (int kk = 0; kk < 16; ++kk) {
            int kb = kk * 4 + lh * 2;
            v2f a, bf;
            a.x  = Qf[tm * 16 + lm][kb + 0];
            a.y  = Qf[tm * 16 + lm][kb + 1];
            bf.x = Kf[tn * 16 + lm][kb + 0];   // B(K=d,N=j) = Kf[j][d]
            bf.y = Kf[tn * 16 + lm][kb + 1];
            acc = wmma_k4(a, bf, acc);
        }
        for (int r = 0; r < 8; ++r) {
            int lrow = tm * 16 + r + lh * 8;
            int jcol = tn * 16 + lm;
            Asc[lrow][jcol] = (jcol <= lrow) ? acc[r] : 0.0f;
        }
    }
    __syncthreads();

    // ---- O = A*V + Q'*KV_prev (32 WMMAs on one accumulator) ----
    v8f o = {};
    for (int kk = 0; kk < 16; ++kk) {
        int kb = kk * 4 + lh * 2;
        v2f a, bf;
        a.x  = Asc[tm * 16 + lm][kb + 0];
        a.y  = Asc[tm * 16 + lm][kb + 1];
        bf.x = Vv[kb + 0][tn * 16 + lm];
        bf.y = Vv[kb + 1][tn * 16 + lm];
        o = wmma_k4(a, bf, o);
    }
    for (int kk = 0; kk < 16; ++kk) {
        int kb = kk * 4 + lh * 2;
        v2f a, bf;
        a.x  = Qf[tm * 16 + lm][kb + 0];
        a.y  = Qf[tm * 16 + lm][kb + 1];
        bf.x = KVp[kb + 0][tn * 16 + lm];
        bf.y = KVp[kb + 1][tn * 16 + lm];
        o = wmma_k4(a, bf, o);
    }

    // ---- normalizer ----
    if (tid < C_) {
        float s = EPS_;
        for (int j = 0; j < C_; ++j) s += Asc[tid][j];          // intra part
        float qd = 0.0f;
        for (int d = 0; d < D_; ++d) qd += Qf[tid][d] * ksum[d]; // inter part
        invden[tid] = 1.0f / (s + qd);
    }
    __syncthreads();

    for (int r = 0; r < 8; ++r) {
        int lrow = tm * 16 + r + lh * 8;
        int m = tn * 16 + lm;
        Out[base + (size_t)lrow * (H_ * D_) + m] = o[r] * invden[lrow];
    }
}

// ---------------------------------------------------------------------------
extern "C" void kernel_launch(void* const* d_in, const int* in_sizes, int n_in,
                              void* d_out, int out_size, void* d_ws, size_t ws_size,
                              hipStream_t stream) {
    (void)in_sizes; (void)n_in; (void)out_size; (void)ws_size;
    const float* Q = (const float*)d_in[0];
    const float* K = (const float*)d_in[1];
    const float* V = (const float*)d_in[2];
    float* Out = (float*)d_out;

    // workspace layout: [ KV chunks : BH*NC*64*64 f32 ][ Ksum chunks : BH*NC*64 f32 ]
    float* kvbuf   = (float*)d_ws;
    float* ksumbuf = kvbuf + (size_t)BH_ * NC_ * D_ * D_;

    dim3 grid(NC_, H_, B_);
    kv_chunk_kernel<<<grid, 512, 0, stream>>>(K, V, kvbuf, ksumbuf);
    prefix_kernel<<<BH_, 256, 0, stream>>>(kvbuf, ksumbuf);
    out_kernel<<<grid, 512, 0, stream>>>(Q, K, V, kvbuf, ksumbuf, Out);
}